// HistoryAttention_51084341019170
// MI455X (gfx1250) — compile-verified
//
#include <hip/hip_runtime.h>

typedef float v2f __attribute__((ext_vector_type(2)));
typedef float v4f __attribute__((ext_vector_type(4)));
typedef float v8f __attribute__((ext_vector_type(8)));

#define B_   128
#define H_   2048
#define E_   1024
#define D_   512
#define EPSF 1e-8f
#define INV_SQRT_D 0.04419417382415922f   // 1/sqrt(512)

// ---------------------------------------------------------------------------
// q = current @ Wq^T + bq            [B_, D_]
// One wave (32 threads) computes one 16x16 output tile via V_WMMA_F32_16X16X4_F32.
// A lane layout: M = lane%16, VGPR j holds K = j + 2*(lane/16)
// B lane layout: N = lane%16, VGPR j holds K = j + 2*(lane/16)
// C/D layout:    VGPR j holds M = j + 8*(lane/16), N = lane%16
// ---------------------------------------------------------------------------
__global__ __launch_bounds__(32) void k_proj_q(const float* __restrict__ cur,
                                               const float* __restrict__ Wq,
                                               const float* __restrict__ bq,
                                               float* __restrict__ q) {
    const int lane = threadIdx.x;
    const int half = lane >> 4;
    const int l16  = lane & 15;
    const int m0 = blockIdx.x * 16;           // batch tile
    const int n0 = blockIdx.y * 16;           // D tile
    const float* arow = cur + (m0 + l16) * E_;            // A[M][.]
    const float* brow = Wq  + (n0 + l16) * E_;            // B[K][N] = Wq[N][K]
    v8f c = {};
    for (int k = 0; k < E_; k += 4) {
        v2f a = *(const v2f*)(arow + k + 2 * half);       // K = k+2h, k+2h+1
        v2f b = *(const v2f*)(brow + k + 2 * half);
        c = __builtin_amdgcn_wmma_f32_16x16x4_f32(false, a, false, b,
                                                  (short)0, c, false, false);
    }
    const float bias = bq[n0 + l16];
#pragma unroll
    for (int j = 0; j < 8; ++j)
        q[(m0 + j + 8 * half) * D_ + n0 + l16] = c[j] + bias;
}

// ---------------------------------------------------------------------------
// qk = q @ Wk                        [B_, E_]   (qk[b] = Wk^T q[b])
// ---------------------------------------------------------------------------
__global__ __launch_bounds__(32) void k_proj_qk(const float* __restrict__ q,
                                                const float* __restrict__ Wk,
                                                float* __restrict__ qk) {
    const int lane = threadIdx.x;
    const int half = lane >> 4;
    const int l16  = lane & 15;
    const int m0 = blockIdx.x * 16;           // batch tile
    const int n0 = blockIdx.y * 16;           // E tile
    const float* arow = q + (m0 + l16) * D_;
    v8f c = {};
    for (int k = 0; k < D_; k += 4) {
        v2f a = *(const v2f*)(arow + k + 2 * half);
        v2f b;
        b.x = Wk[(k + 0 + 2 * half) * E_ + n0 + l16];     // B[K][N] = Wk[K*E + N]
        b.y = Wk[(k + 1 + 2 * half) * E_ + n0 + l16];
        c = __builtin_amdgcn_wmma_f32_16x16x4_f32(false, a, false, b,
                                                  (short)0, c, false, false);
    }
#pragma unroll
    for (int j = 0; j < 8; ++j)
        qk[(m0 + j + 8 * half) * E_ + n0 + l16] = c[j];
}

// ---------------------------------------------------------------------------
// Per-batch scalars: qdotbk[b] = q[b].bk ;  ncur[b] = max(||cur[b]||, eps)
// One wave per batch row, wave32 shuffle reduction.
// ---------------------------------------------------------------------------
__global__ __launch_bounds__(32) void k_prep(const float* __restrict__ q,
                                             const float* __restrict__ bk,
                                             const float* __restrict__ cur,
                                             float* __restrict__ qdotbk,
                                             float* __restrict__ ncur) {
    const int b = blockIdx.x;
    const int lane = threadIdx.x;
    float s = 0.f, n = 0.f;
    for (int d = lane; d < D_; d += 32) s += q[b * D_ + d] * bk[d];
    for (int e = lane; e < E_; e += 32) { float v = cur[b * E_ + e]; n += v * v; }
#pragma unroll
    for (int off = 16; off > 0; off >>= 1) {
        s += __shfl_xor(s, off, 32);
        n += __shfl_xor(n, off, 32);
    }
    if (lane == 0) {
        qdotbk[b] = s;
        ncur[b]   = fmaxf(sqrtf(n), EPSF);
    }
}

// ---------------------------------------------------------------------------
// Dominant streaming pass: 1 GiB of hist, one wave per (b,h) row.
// Three fused reductions per row; hist loaded non-temporally (streamed once).
// ---------------------------------------------------------------------------
__global__ __launch_bounds__(256) void k_main(const float* __restrict__ hist,
                                              const float* __restrict__ qk,
                                              const float* __restrict__ cur,
                                              const float* __restrict__ qdotbk,
                                              const float* __restrict__ ncur,
                                              const float* __restrict__ w_cos,
                                              const float* __restrict__ w_att,
                                              float* __restrict__ out) {
    const int lane = threadIdx.x & 31;
    const int wid  = threadIdx.x >> 5;
    const int row  = blockIdx.x * 8 + wid;     // row in [0, B*H)
    const int b    = row >> 11;                // H = 2048
    const float* hrow = hist + (size_t)row * E_;
    const float* qrow = qk  + b * E_;
    const float* crow = cur + b * E_;
    float dq = 0.f, dc = 0.f, ss = 0.f;
#pragma unroll
    for (int i = 0; i < 8; ++i) {
        const int o = (lane + 32 * i) * 4;     // coalesced float4 per lane
        v4f hv = __builtin_nontemporal_load((const v4f*)(hrow + o));
        v4f qv = *(const v4f*)(qrow + o);
        v4f cv = *(const v4f*)(crow + o);
        dq += hv.x * qv.x + hv.y * qv.y + hv.z * qv.z + hv.w * qv.w;
        dc += hv.x * cv.x + hv.y * cv.y + hv.z * cv.z + hv.w * cv.w;
        ss += hv.x * hv.x + hv.y * hv.y + hv.z * hv.z + hv.w * hv.w;
    }
#pragma unroll
    for (int off = 16; off > 0; off >>= 1) {
        dq += __shfl_xor(dq, off, 32);
        dc += __shfl_xor(dc, off, 32);
        ss += __shfl_xor(ss, off, 32);
    }
    if (lane == 0) {
        const float score = (dq + qdotbk[b]) * INV_SQRT_D;
        const float cosv  = dc / (ncur[b] * fmaxf(sqrtf(ss), EPSF));
        out[row] = w_att[0] * score + w_cos[0] * cosv;
    }
}

extern "C" void kernel_launch(void* const* d_in, const int* in_sizes, int n_in,
                              void* d_out, int out_size, void* d_ws, size_t ws_size,
                              hipStream_t stream) {
    const float* cur   = (const float*)d_in[0];   // [B,E]
    const float* hist  = (const float*)d_in[1];   // [B,H,E]
    const float* Wq    = (const float*)d_in[2];   // [D,E]
    const float* bq    = (const float*)d_in[3];   // [D]
    const float* Wk    = (const float*)d_in[4];   // [D,E]
    const float* bk    = (const float*)d_in[5];   // [D]
    const float* w_cos = (const float*)d_in[6];   // [1]
    const float* w_att = (const float*)d_in[7];   // [1]
    float* out = (float*)d_out;                   // [B,H]

    float* ws    = (float*)d_ws;
    float* q     = ws;                     // B_*D_   = 65536 floats
    float* qk    = q  + B_ * D_;           // B_*E_   = 131072 floats
    float* qdbk  = qk + B_ * E_;           // B_ floats
    float* ncurv = qdbk + B_;              // B_ floats

    k_proj_q <<<dim3(B_ / 16, D_ / 16), 32, 0, stream>>>(cur, Wq, bq, q);
    k_proj_qk<<<dim3(B_ / 16, E_ / 16), 32, 0, stream>>>(q, Wk, qk);
    k_prep   <<<B_, 32, 0, stream>>>(q, bk, cur, qdbk, ncurv);
    k_main   <<<(B_ * H_) / 8, 256, 0, stream>>>(hist, qk, cur, qdbk, ncurv,
                                                 w_cos, w_att, out);
}